// GeodesicLoss_40475771797581
// MI455X (gfx1250) — compile-verified
//
#include <hip/hip_runtime.h>

// Reference constants
#define GEO_ALPHA      1.7f
#define GEO_BETA       0.01f
#define GEO_DIST_CLAMP 0.1f

typedef float v2f __attribute__((ext_vector_type(2)));
typedef float v8f __attribute__((ext_vector_type(8)));
typedef int   v4i __attribute__((ext_vector_type(4)));

// Half-wave f32 sum via V_WMMA_F32_16X16X4_F32.
// A (16x4 f32): per-lane partial in slot 0, zero in slot 1. B = ones, C = 0.
// D[m][n] = sum_k A[m][k]; columns identical. Per the 16x16 f32 C/D layout,
// lanes 0-15 hold rows 0-7 in c[0..7] and lanes 16-31 hold rows 8-15, so
// sum(c[0..7]) on lane 0 is the sum of lanes 0-7,16-23 and on lane 16 of lanes 8-15,24-31.
__device__ __forceinline__ float wave_halfsum_wmma(float x) {
    v2f a; a[0] = x;    a[1] = 0.0f;
    v2f b; b[0] = 1.0f; b[1] = 1.0f;
    v8f c = {};
    c = __builtin_amdgcn_wmma_f32_16x16x4_f32(
        /*neg_a=*/false, a, /*neg_b=*/false, b,
        /*c_mod=*/(short)0, c, /*reuse_a=*/false, /*reuse_b=*/false);
    return ((c[0] + c[1]) + (c[2] + c[3])) + ((c[4] + c[5]) + (c[6] + c[7]));
}

// Zero the scalar accumulator; optionally precompute per-node covalent radius.
__global__ void geo_prep_kernel(const int* __restrict__ an,
                                const float* __restrict__ cov,
                                float* __restrict__ node_r,
                                int n_nodes,
                                float* __restrict__ out) {
    int i = blockIdx.x * blockDim.x + threadIdx.x;
    if (i == 0) out[0] = 0.0f;
    if (node_r != nullptr) {
        int stride = gridDim.x * blockDim.x;
        for (int v = i; v < n_nodes; v += stride)
            node_r[v] = cov[an[v]];
    }
}

// Per-edge squared JVP term. Exactly two IEEE divides (1/r, 1/dist).
template <bool USE_NR>
__device__ __forceinline__ float edge_term(int s, int d,
                                           const float* __restrict__ x,
                                           const float* __restrict__ dx,
                                           const float* __restrict__ node_r,
                                           const int*   __restrict__ an,
                                           const float* __restrict__ cov) {
    float r;
    if (USE_NR) r = node_r[s] + node_r[d];
    else        r = cov[an[s]] + cov[an[d]];

    const float* xs = x  + 3 * (size_t)s;
    const float* xd = x  + 3 * (size_t)d;
    const float* ts = dx + 3 * (size_t)s;
    const float* td = dx + 3 * (size_t)d;

    float df0 = xs[0] - xd[0];
    float df1 = xs[1] - xd[1];
    float df2 = xs[2] - xd[2];
    float dist2 = df0 * df0 + df1 * df1 + df2 * df2;
    float dist  = sqrtf(dist2);

    float tv0 = ts[0] - td[0];
    float tv1 = ts[1] - td[1];
    float tv2 = ts[2] - td[2];
    float ddot = df0 * tv0 + df1 * tv1 + df2 * tv2;

    float inv_r    = 1.0f / r;
    float inv_dist = 1.0f / dist;

    // d(coord)/d(dist)
    float g = (-GEO_ALPHA * inv_r) * expf(-GEO_ALPHA * (dist - r) * inv_r);
    if (dist > GEO_DIST_CLAMP)
        g -= (GEO_BETA * r) * inv_dist * inv_dist;

    // jvp = g * (diff . tdiff) / dist
    float jvp = g * ddot * inv_dist;
    return jvp * jvp;
}

template <bool USE_NR>
__global__ void geo_edge_kernel(const float* __restrict__ x,
                                const float* __restrict__ dx,
                                const int*   __restrict__ src,
                                const int*   __restrict__ dst,
                                const float* __restrict__ node_r,
                                const int*   __restrict__ an,
                                const float* __restrict__ cov,
                                const int*   __restrict__ n_graphs_p,
                                float*       __restrict__ out,
                                int n_edges) {
    float acc = 0.0f;
    const int tid    = blockIdx.x * blockDim.x + threadIdx.x;
    const int stride = gridDim.x * blockDim.x;

    // Vector body: 4 edges per iteration, 128-bit nontemporal index loads so
    // the one-pass 102 MB edge stream doesn't evict the L2-resident node data.
    const int ngroups = n_edges >> 2;
    const v4i* src4 = (const v4i*)src;
    const v4i* dst4 = (const v4i*)dst;
    for (int g = tid; g < ngroups; g += stride) {
        v4i s4 = __builtin_nontemporal_load(src4 + g);
        v4i d4 = __builtin_nontemporal_load(dst4 + g);
        acc += edge_term<USE_NR>(s4[0], d4[0], x, dx, node_r, an, cov);
        acc += edge_term<USE_NR>(s4[1], d4[1], x, dx, node_r, an, cov);
        acc += edge_term<USE_NR>(s4[2], d4[2], x, dx, node_r, an, cov);
        acc += edge_term<USE_NR>(s4[3], d4[3], x, dx, node_r, an, cov);
    }
    // Scalar tail (<= 3 edges total across the grid).
    for (int e = (ngroups << 2) + tid; e < n_edges; e += stride) {
        int s = __builtin_nontemporal_load(src + e);
        int d = __builtin_nontemporal_load(dst + e);
        acc += edge_term<USE_NR>(s, d, x, dx, node_r, an, cov);
    }

    // Wave reduction on the matrix unit. EXEC is all-1s here (full 256-thread
    // blocks, reconverged after the loops). Lane 0 holds the sum of lanes 0-7,16-23,
    // lane 16 that of lanes 8-15,24-31; each deposits one LDS slot — no permute.
    float h = wave_halfsum_wmma(acc);

    __shared__ float smem[16];
    int lane = threadIdx.x & 31;
    int wid  = threadIdx.x >> 5;
    if ((lane & 15) == 0)
        smem[2 * wid + (lane >> 4)] = h;
    __syncthreads();

    if (threadIdx.x == 0) {
        int nslots = (blockDim.x >> 5) * 2;
        float b = 0.0f;
        for (int i = 0; i < nslots; ++i) b += smem[i];
        float inv_ng = 1.0f / (float)(*n_graphs_p);
        atomicAdd(out, b * inv_ng);  // one atomic per block
    }
}

extern "C" void kernel_launch(void* const* d_in, const int* in_sizes, int n_in,
                              void* d_out, int out_size, void* d_ws, size_t ws_size,
                              hipStream_t stream) {
    // Input order per setup_inputs():
    // 0: x_t f32 [N,3], 1: dx_dt f32 [N,3], 2: edge_index i32 [2,E],
    // 3: atomic_numbers i32 [N], 4: batch_ids i32 [N] (unused — the final mean
    //    collapses the segment sum to a global sum / n_graphs),
    // 5: covalent_radii f32 [119], 6: n_graphs i32 [1]
    const float* x   = (const float*)d_in[0];
    const float* dx  = (const float*)d_in[1];
    const int*   ei  = (const int*)d_in[2];
    const int*   an  = (const int*)d_in[3];
    const float* cov = (const float*)d_in[5];
    const int*   ngp = (const int*)d_in[6];
    float*       out = (float*)d_out;

    const int n_edges = in_sizes[2] / 2;
    const int n_nodes = in_sizes[3];
    const int* src = ei;
    const int* dst = ei + n_edges;

    // Per-node radius in scratch if it fits (1.6 MB for 400k nodes): turns the
    // dependent an->cov gather chain into a single L2-resident gather.
    float* node_r = nullptr;
    if (ws_size >= (size_t)n_nodes * sizeof(float))
        node_r = (float*)d_ws;

    int prep_blocks = node_r ? ((n_nodes + 255) / 256 < 2048 ? (n_nodes + 255) / 256 : 2048) : 1;
    geo_prep_kernel<<<prep_blocks, 256, 0, stream>>>(an, cov, node_r, n_nodes, out);

    // int4 index loads need 16B alignment of both stream halves.
    bool aligned4 = ((n_edges & 3) == 0) && ((((uintptr_t)ei) & 15) == 0);

    const int blocks = 3072, threads = 256;  // ~786k threads to hide L2 gather latency
    if (node_r && aligned4)
        geo_edge_kernel<true ><<<blocks, threads, 0, stream>>>(x, dx, src, dst, node_r, an, cov, ngp, out, n_edges);
    else if (node_r)
        geo_edge_kernel<true ><<<blocks, threads, 0, stream>>>(x, dx, src, dst, node_r, an, cov, ngp, out, n_edges);
    else
        geo_edge_kernel<false><<<blocks, threads, 0, stream>>>(x, dx, src, dst, node_r, an, cov, ngp, out, n_edges);
}